// MaterialPointModel3d_34703335752435
// MI455X (gfx1250) — compile-verified
//
#include <hip/hip_runtime.h>
#include <math.h>

// ---------------------------------------------------------------------------
// MPM substep (MLS-MPM, quadratic B-spline, 3 materials) for gfx1250.
// Grid state lives in d_ws as interleaved float4 {vx, vy, vz, m} per node:
// 128^3 nodes * 16 B = 32 MiB (fits in the MI455X 192 MB L2 -> scatter and
// gather traffic never leaves L2).  Pipeline:
//   zero grid -> P2G (atomic scatter) -> grid update (async-LDS staged RMW)
//   -> G2P gather.
// ---------------------------------------------------------------------------

namespace {
constexpr int   kNGrid = 128;
constexpr int   kN3    = kNGrid * kNGrid * kNGrid;
constexpr float kDx    = 1.0f / (float)kNGrid;
constexpr float kInvDx = (float)kNGrid;
constexpr float kDt    = 1.0e-4f / 2.0f;                       // QUALITY = 2
constexpr float kPVol  = (kDx * 0.5f) * (kDx * 0.5f) * (kDx * 0.5f);
constexpr float kPMass = kPVol * 1.0f;
constexpr float kMu0   = 5000.0f / (2.0f * (1.0f + 0.2f));
constexpr float kLam0  = 5000.0f * 0.2f / ((1.0f + 0.2f) * (1.0f - 2.0f * 0.2f));
constexpr float kStressScale = -kDt * kPVol * 4.0f * kInvDx * kInvDx;
} // namespace

// ---------------------------------------------------------------------------
// 3x3 SVD via cyclic Jacobi on S = A^T A.  Returns A = U * diag(sig) * V^T
// with sig >= 0 (LAPACK/jnp convention).  Order of singular values does not
// matter for this model (clip / prod are order-invariant).
// ---------------------------------------------------------------------------
__device__ __forceinline__ void jacobi_rot(float S[3][3], float V[3][3],
                                           int p, int q) {
  float apq = S[p][q];
  if (fabsf(apq) < 1.0e-20f) return;
  float tau = (S[q][q] - S[p][p]) / (2.0f * apq);
  float t   = copysignf(1.0f, tau) / (fabsf(tau) + sqrtf(1.0f + tau * tau));
  float c   = rsqrtf(1.0f + t * t);
  float s   = t * c;
#pragma unroll
  for (int k = 0; k < 3; ++k) {            // S <- S * G
    float skp = S[k][p], skq = S[k][q];
    S[k][p] = c * skp - s * skq;
    S[k][q] = s * skp + c * skq;
  }
#pragma unroll
  for (int k = 0; k < 3; ++k) {            // S <- G^T * S
    float spk = S[p][k], sqk = S[q][k];
    S[p][k] = c * spk - s * sqk;
    S[q][k] = s * spk + c * sqk;
  }
#pragma unroll
  for (int k = 0; k < 3; ++k) {            // V <- V * G
    float vkp = V[k][p], vkq = V[k][q];
    V[k][p] = c * vkp - s * vkq;
    V[k][q] = s * vkp + c * vkq;
  }
}

__device__ __forceinline__ void svd3(const float A[3][3], float U[3][3],
                                     float sig[3], float V[3][3]) {
  float S[3][3];
#pragma unroll
  for (int i = 0; i < 3; ++i)
#pragma unroll
    for (int j = 0; j < 3; ++j) {
      float acc = 0.0f;
#pragma unroll
      for (int k = 0; k < 3; ++k) acc += A[k][i] * A[k][j];
      S[i][j] = acc;
      V[i][j] = (i == j) ? 1.0f : 0.0f;
    }
#pragma unroll
  for (int sweep = 0; sweep < 6; ++sweep) {
    jacobi_rot(S, V, 0, 1);
    jacobi_rot(S, V, 0, 2);
    jacobi_rot(S, V, 1, 2);
  }
#pragma unroll
  for (int i = 0; i < 3; ++i) sig[i] = sqrtf(fmaxf(S[i][i], 0.0f));
#pragma unroll
  for (int i = 0; i < 3; ++i) {            // U[:,i] = A * V[:,i] / sig[i]
    float inv = 1.0f / fmaxf(sig[i], 1.0e-12f);
#pragma unroll
    for (int r = 0; r < 3; ++r) {
      float acc = 0.0f;
#pragma unroll
      for (int k = 0; k < 3; ++k) acc += A[r][k] * V[k][i];
      U[r][i] = acc * inv;
    }
  }
}

// ---------------------------------------------------------------------------
// Kernel 0: zero the grid (harness poisons d_ws; must re-zero every call).
// ---------------------------------------------------------------------------
__global__ void __launch_bounds__(256) mpm_zero_grid(float4* __restrict__ grid,
                                                     int n4) {
  int i = blockIdx.x * blockDim.x + threadIdx.x;
  if (i < n4) grid[i] = make_float4(0.0f, 0.0f, 0.0f, 0.0f);
}

// ---------------------------------------------------------------------------
// Kernel 1: particle constitutive update + P2G scatter.
// Writes updated F (cols 15..23) and Jp (col 24) into d_out.
// ---------------------------------------------------------------------------
__global__ void __launch_bounds__(256) mpm_p2g(
    const float* __restrict__ xin, const float* __restrict__ vin,
    const float* __restrict__ Cin, const float* __restrict__ Fin,
    const float* __restrict__ Jpin, const int* __restrict__ matin,
    float* __restrict__ grid, float* __restrict__ out, int npart) {
  int p = blockIdx.x * blockDim.x + threadIdx.x;
  if (p >= npart) return;

  float xp[3], vp[3], Cm[3][3], Fm[3][3];
#pragma unroll
  for (int d = 0; d < 3; ++d) {
    xp[d] = xin[p * 3 + d];
    vp[d] = vin[p * 3 + d];
  }
#pragma unroll
  for (int i = 0; i < 3; ++i)
#pragma unroll
    for (int j = 0; j < 3; ++j) {
      Cm[i][j] = Cin[p * 9 + i * 3 + j];
      Fm[i][j] = Fin[p * 9 + i * 3 + j];
    }
  float Jp = Jpin[p];
  int   m  = matin[p];

  // base / fx / quadratic B-spline weights  w[node][dim]
  int   base[3];
  float fx[3], w[3][3];
#pragma unroll
  for (int d = 0; d < 3; ++d) {
    float xs = xp[d] * kInvDx;
    float bf = floorf(xs - 0.5f);
    base[d]  = (int)bf;
    fx[d]    = xs - bf;
    float a = 1.5f - fx[d];
    float b = fx[d] - 1.0f;
    float c = fx[d] - 0.5f;
    w[0][d] = 0.5f * a * a;
    w[1][d] = 0.75f - b * b;
    w[2][d] = 0.5f * c * c;
  }

  // F <- (I + dt*C) @ F
  float Fn[3][3];
#pragma unroll
  for (int i = 0; i < 3; ++i)
#pragma unroll
    for (int j = 0; j < 3; ++j) {
      float acc = Fm[i][j];
#pragma unroll
      for (int k = 0; k < 3; ++k) acc += kDt * Cm[i][k] * Fm[k][j];
      Fn[i][j] = acc;
    }

  // hardening
  float h = expf(10.0f * (1.0f - Jp));
  h = fminf(fmaxf(h, 0.1f), 5.0f);
  if (m == 1) h = 0.3f;                       // jelly
  float mu = (m == 0) ? 0.0f : kMu0 * h;      // liquid: mu = 0
  float la = kLam0 * h;

  // SVD + plasticity
  float U[3][3], V[3][3], sig[3];
  svd3(Fn, U, sig, V);

  float nsig[3];
#pragma unroll
  for (int i = 0; i < 3; ++i) {
    float s = sig[i];
    nsig[i] = (m == 2) ? fminf(fmaxf(s, 1.0f - 2.5e-2f), 1.0f + 4.5e-3f) : s;
  }
  if (m == 2)
    Jp *= (sig[0] / nsig[0]) * (sig[1] / nsig[1]) * (sig[2] / nsig[2]);
  float J = nsig[0] * nsig[1] * nsig[2];

  float Ff[3][3];
  if (m == 0) {
    float sj = sqrtf(J);
#pragma unroll
    for (int i = 0; i < 3; ++i)
#pragma unroll
      for (int j = 0; j < 3; ++j) Ff[i][j] = (i == j) ? sj : 0.0f;
  } else if (m == 2) {
#pragma unroll
    for (int i = 0; i < 3; ++i)
#pragma unroll
      for (int j = 0; j < 3; ++j) {
        float acc = 0.0f;
#pragma unroll
        for (int k = 0; k < 3; ++k) acc += U[i][k] * nsig[k] * V[j][k];
        Ff[i][j] = acc;
      }
  } else {
#pragma unroll
    for (int i = 0; i < 3; ++i)
#pragma unroll
      for (int j = 0; j < 3; ++j) Ff[i][j] = Fn[i][j];
  }

  // R = U V^T ; stress = 2 mu (F - R) F^T + la J (J-1) I ; affine
  float diag = la * J * (J - 1.0f);
  float affine[3][3];
#pragma unroll
  for (int i = 0; i < 3; ++i)
#pragma unroll
    for (int j = 0; j < 3; ++j) {
      float acc = 0.0f;
#pragma unroll
      for (int k = 0; k < 3; ++k) {
        float Rik = U[i][0] * V[k][0] + U[i][1] * V[k][1] + U[i][2] * V[k][2];
        acc += (Ff[i][k] - Rik) * Ff[j][k];
      }
      float st = 2.0f * mu * acc + ((i == j) ? diag : 0.0f);
      affine[i][j] = kStressScale * st + kPMass * Cm[i][j];
    }

  // write F, Jp
  float* orow = out + (size_t)p * 25;
#pragma unroll
  for (int i = 0; i < 3; ++i)
#pragma unroll
    for (int j = 0; j < 3; ++j) orow[15 + i * 3 + j] = Ff[i][j];
  orow[24] = Jp;

  // P2G scatter: 27 nodes x (3 momentum + 1 mass) float atomics into L2.
  // Linear node index matches the reference's lin arithmetic exactly.
  float pmv[3];
#pragma unroll
  for (int d = 0; d < 3; ++d) pmv[d] = kPMass * vp[d];

  int lin_i = (base[0] * kNGrid + base[1]) * kNGrid + base[2];
  for (int i = 0; i < 3; ++i, lin_i += kNGrid * kNGrid) {
    float dp0 = ((float)i - fx[0]) * kDx;
    float a0 = pmv[0] + affine[0][0] * dp0;
    float a1 = pmv[1] + affine[1][0] * dp0;
    float a2 = pmv[2] + affine[2][0] * dp0;
    int lin_j = lin_i;
    for (int j = 0; j < 3; ++j, lin_j += kNGrid) {
      float dp1 = ((float)j - fx[1]) * kDx;
      float b0 = a0 + affine[0][1] * dp1;
      float b1 = a1 + affine[1][1] * dp1;
      float b2 = a2 + affine[2][1] * dp1;
      float wij = w[i][0] * w[j][1];
      for (int k = 0; k < 3; ++k) {
        float weight = wij * w[k][2];
        float dp2 = ((float)k - fx[2]) * kDx;
        int node = lin_j + k;
        if ((unsigned)node < (unsigned)kN3) {
          float* gp = grid + (size_t)node * 4;
          atomicAdd(gp + 0, weight * (b0 + affine[0][2] * dp2));
          atomicAdd(gp + 1, weight * (b1 + affine[1][2] * dp2));
          atomicAdd(gp + 2, weight * (b2 + affine[2][2] * dp2));
          atomicAdd(gp + 3, weight * kPMass);
        }
      }
    }
  }
}

// ---------------------------------------------------------------------------
// Kernel 2: grid normalize + gravity + boundary clamp (in place).
// Each block owns 256 consecutive nodes (4 KB).  The per-lane 16 B node is
// staged into LDS with the gfx1250 async-DMA path (GLOBAL_LOAD_ASYNC_TO_LDS,
// tracked by ASYNCcnt, no VGPR destination), the lane overlaps the copy with
// node-coordinate / boundary-mask VALU work, drains with s_wait_asynccnt 0,
// and reads its own staged element back (no barrier: self-consumption only).
// ---------------------------------------------------------------------------
__global__ void __launch_bounds__(256) mpm_grid_update(
    float4* __restrict__ grid, const float* __restrict__ grav) {
  __shared__ float4 tile[256];
  int g = blockIdx.x * 256 + threadIdx.x;
  if (g >= kN3) return;  // kN3 % 256 == 0 -> never taken, EXEC stays full

  // Kick off the async global->LDS copy of this lane's node (16 bytes).
  unsigned lds_off = (unsigned)(size_t)(void*)&tile[threadIdx.x];
  unsigned long long gaddr = (unsigned long long)(grid + g);
  asm volatile("global_load_async_to_lds_b128 %0, %1, off"
               :
               : "v"(lds_off), "v"(gaddr)
               : "memory");

  // Overlap: gravity scale + node coordinates + boundary masks.
  float gx = grav[0], gy = grav[1], gz = grav[2];
  const float gscale = kDt * 30.0f;
  int gi = g / (kNGrid * kNGrid);
  int gj = (g / kNGrid) % kNGrid;
  int gk = g % kNGrid;

  asm volatile("s_wait_asynccnt 0x0" ::: "memory");
  float4 gv = tile[threadIdx.x];

  float mass = gv.w;
  float vx = 0.0f, vy = 0.0f, vz = 0.0f;
  if (mass > 0.0f) {
    float inv = 1.0f / fmaxf(mass, 1.0e-30f);
    vx = gv.x * inv + gscale * gx;
    vy = gv.y * inv + gscale * gy;
    vz = gv.z * inv + gscale * gz;
  }
  if (gi < 3 && vx < 0.0f) vx = 0.0f;
  if (gi > kNGrid - 3 && vx > 0.0f) vx = 0.0f;
  if (gj < 3 && vy < 0.0f) vy = 0.0f;
  if (gj > kNGrid - 3 && vy > 0.0f) vy = 0.0f;
  if (gk < 3 && vz < 0.0f) vz = 0.0f;
  if (gk > kNGrid - 3 && vz > 0.0f) vz = 0.0f;
  grid[g] = make_float4(vx, vy, vz, mass);
}

// ---------------------------------------------------------------------------
// Kernel 3: G2P gather -> new x, v, C (cols 0..14 of d_out).
// ---------------------------------------------------------------------------
__global__ void __launch_bounds__(256) mpm_g2p(
    const float* __restrict__ xin, const float4* __restrict__ grid,
    float* __restrict__ out, int npart) {
  int p = blockIdx.x * blockDim.x + threadIdx.x;
  if (p >= npart) return;

  float xp[3];
#pragma unroll
  for (int d = 0; d < 3; ++d) xp[d] = xin[p * 3 + d];

  int   base[3];
  float fx[3], w[3][3];
#pragma unroll
  for (int d = 0; d < 3; ++d) {
    float xs = xp[d] * kInvDx;
    float bf = floorf(xs - 0.5f);
    base[d]  = (int)bf;
    fx[d]    = xs - bf;
    float a = 1.5f - fx[d];
    float b = fx[d] - 1.0f;
    float c = fx[d] - 0.5f;
    w[0][d] = 0.5f * a * a;
    w[1][d] = 0.75f - b * b;
    w[2][d] = 0.5f * c * c;
  }

  float nv[3] = {0.0f, 0.0f, 0.0f};
  float nC[3][3] = {{0.f, 0.f, 0.f}, {0.f, 0.f, 0.f}, {0.f, 0.f, 0.f}};

  int lin_i = (base[0] * kNGrid + base[1]) * kNGrid + base[2];
  for (int i = 0; i < 3; ++i, lin_i += kNGrid * kNGrid) {
    float dp0 = (float)i - fx[0];
    int lin_j = lin_i;
    for (int j = 0; j < 3; ++j, lin_j += kNGrid) {
      float dp1 = (float)j - fx[1];
      float wij = w[i][0] * w[j][1];
      for (int k = 0; k < 3; ++k) {
        float weight = wij * w[k][2];
        float dp2 = (float)k - fx[2];
        int node = lin_j + k;
        float4 gv = make_float4(0.f, 0.f, 0.f, 0.f);
        if ((unsigned)node < (unsigned)kN3) gv = grid[node];  // b128 load
        float wx = weight * gv.x, wy = weight * gv.y, wz = weight * gv.z;
        nv[0] += wx;
        nv[1] += wy;
        nv[2] += wz;
        nC[0][0] += wx * dp0;
        nC[0][1] += wx * dp1;
        nC[0][2] += wx * dp2;
        nC[1][0] += wy * dp0;
        nC[1][1] += wy * dp1;
        nC[1][2] += wy * dp2;
        nC[2][0] += wz * dp0;
        nC[2][1] += wz * dp1;
        nC[2][2] += wz * dp2;
      }
    }
  }

  float* orow = out + (size_t)p * 25;
#pragma unroll
  for (int d = 0; d < 3; ++d) {
    orow[d]     = xp[d] + kDt * nv[d];  // new x
    orow[3 + d] = nv[d];                // new v
  }
  float cs = 4.0f * kInvDx;
#pragma unroll
  for (int i = 0; i < 3; ++i)
#pragma unroll
    for (int j = 0; j < 3; ++j) orow[6 + i * 3 + j] = cs * nC[i][j];
}

// ---------------------------------------------------------------------------
extern "C" void kernel_launch(void* const* d_in, const int* in_sizes, int n_in,
                              void* d_out, int out_size, void* d_ws,
                              size_t ws_size, hipStream_t stream) {
  const float* x    = (const float*)d_in[0];
  const float* v    = (const float*)d_in[1];
  const float* C    = (const float*)d_in[2];
  const float* F    = (const float*)d_in[3];
  const float* Jp   = (const float*)d_in[4];
  const int*   mat  = (const int*)d_in[5];
  const float* grav = (const float*)d_in[6];
  float*       out  = (float*)d_out;
  float*       grid = (float*)d_ws;        // needs 4 * 128^3 floats = 32 MiB
  const int npart   = in_sizes[4];         // Jp has one element per particle

  const int threads = 256;
  const int pblocks = (npart + threads - 1) / threads;
  const int gblocks = (kN3 + threads - 1) / threads;

  mpm_zero_grid<<<gblocks, threads, 0, stream>>>((float4*)grid, kN3);
  mpm_p2g<<<pblocks, threads, 0, stream>>>(x, v, C, F, Jp, mat, grid, out,
                                           npart);
  mpm_grid_update<<<gblocks, threads, 0, stream>>>((float4*)grid, grav);
  mpm_g2p<<<pblocks, threads, 0, stream>>>(x, (const float4*)grid, out, npart);
}